// MotionSSM_3882650437060
// MI455X (gfx1250) — compile-verified
//
#include <hip/hip_runtime.h>
#include <hip/hip_bf16.h>
#include <math.h>

// ---------------------------------------------------------------------------
// MotionSSM (4-layer Mamba) for MI455X / gfx1250, wave32, f32 WMMA 16x16x4,
// LDS-staged double-buffered GEMM.
// ---------------------------------------------------------------------------

#define NLAYERS 4
#define DMODEL 512
#define DSTATE 32
#define DINNER 1024
#define DCONV 4
#define DTRANK 32
#define BATCH 8
#define TLEN 512
#define NROWS (BATCH * TLEN)   // 4096 token rows

typedef __attribute__((ext_vector_type(2))) float v2f;
typedef __attribute__((ext_vector_type(8))) float v8f;

enum { ACT_NONE = 0, ACT_SOFTPLUS = 1 };

__device__ __forceinline__ float softplus_f(float v) {
    return fmaxf(v, 0.f) + log1pf(__expf(-fabsf(v)));
}
__device__ __forceinline__ float silu_f(float v) {
    return v * (1.f / (1.f + __expf(-v)));
}

// ---------------------------------------------------------------------------
// Row-major GEMM  C[M,N](ldc) = act(A[M,K](lda) @ W[K,N](ldw) + bias) [+= C]
// Block = 128 threads = 4 waves; block tile 64 x (16*NT); K-chunk = 32.
// A-tile and W-tile staged in LDS (double-buffered via registers); W packed
// in K-pairs so B fragments are single ds_load_b64. WMMA 16x16x4 f32.
// Requires M%64==0, N%(16*NT)==0, K%32==0 (K=32,512,1024 here).
// ---------------------------------------------------------------------------
template <int NT, int ACT, bool ACC>
__global__ __launch_bounds__(128) void gemm_wmma_f32(
    const float* __restrict__ A, int lda,
    const float* __restrict__ W, int ldw,
    const float* __restrict__ bias,
    float* __restrict__ C, int ldc,
    int K)
{
    constexpr int KC = 32;            // K-chunk
    constexpr int BN = 16 * NT;       // block N width
    constexpr int AP = 36;            // padded A row stride (bank-spread, 8B ok)

    __shared__ float sA[64 * AP];               // 64 rows x 32 cols (padded)
    __shared__ float sW[(KC / 2) * BN * 2];     // K-pair packed: [p][n][2]

    const int tid   = threadIdx.x;
    const int lane  = tid & 31;
    const int wave  = tid >> 5;
    const int rowA  = blockIdx.y * 64;
    const int nbase = blockIdx.x * BN;

    // fragment coordinates (16x4 f32 WMMA layout)
    const int m    = lane & 15;                 // A/B fragment row/col
    const int kgrp = (lane >> 4) << 1;          // 0 or 2

    float4 aReg[4];        // 64*32 floats / 128 threads / 4 = 4 float4 each
    float4 wReg[NT];       // KC*BN floats / 128 threads / 4 = NT float4 each

    auto loadChunk = [&](int k0) {
#pragma unroll
        for (int j = 0; j < 4; ++j) {
            const int idx = j * 512 + tid * 4;
            const int r = idx >> 5, c = idx & 31;
            aReg[j] = *reinterpret_cast<const float4*>(
                &A[(size_t)(rowA + r) * lda + k0 + c]);
        }
#pragma unroll
        for (int j = 0; j < NT; ++j) {
            const int idx = j * 512 + tid * 4;
            const int k = idx / BN, n = idx % BN;
            wReg[j] = *reinterpret_cast<const float4*>(
                &W[(size_t)(k0 + k) * ldw + nbase + n]);
        }
    };
    auto storeChunk = [&]() {
#pragma unroll
        for (int j = 0; j < 4; ++j) {
            const int idx = j * 512 + tid * 4;
            const int r = idx >> 5, c = idx & 31;
            *reinterpret_cast<float4*>(&sA[r * AP + c]) = aReg[j];
        }
#pragma unroll
        for (int j = 0; j < NT; ++j) {
            const int idx = j * 512 + tid * 4;
            const int k = idx / BN, n = idx % BN;
            const int p = k >> 1, lo = k & 1;
            sW[(p * BN + n + 0) * 2 + lo] = wReg[j].x;
            sW[(p * BN + n + 1) * 2 + lo] = wReg[j].y;
            sW[(p * BN + n + 2) * 2 + lo] = wReg[j].z;
            sW[(p * BN + n + 3) * 2 + lo] = wReg[j].w;
        }
    };

    v8f acc[NT] = {};

    loadChunk(0);
    for (int k0 = 0; k0 < K; k0 += KC) {
        storeChunk();
        __syncthreads();
        if (k0 + KC < K) loadChunk(k0 + KC);   // overlap next global fetch
#pragma unroll
        for (int k = 0; k < KC; k += 4) {
            v2f a = *reinterpret_cast<const v2f*>(
                &sA[(wave * 16 + m) * AP + k + kgrp]);
            const int p = (k + kgrp) >> 1;
#pragma unroll
            for (int t = 0; t < NT; ++t) {
                v2f b = *reinterpret_cast<const v2f*>(
                    &sW[(p * BN + m + t * 16) * 2]);
                acc[t] = __builtin_amdgcn_wmma_f32_16x16x4_f32(
                    false, a, false, b, (short)0, acc[t], false, false);
            }
        }
        __syncthreads();
    }

    // C/D layout: lanes 0-15 -> N=lane, rows M=r ; lanes 16-31 -> rows M=r+8
    const int mr_base = rowA + wave * 16 + ((lane >> 4) << 3);
#pragma unroll
    for (int t = 0; t < NT; ++t) {
        const int n = nbase + t * 16 + m;
        const float bv = bias ? bias[n] : 0.f;
#pragma unroll
        for (int r = 0; r < 8; ++r) {
            float v = acc[t][r] + bv;
            if (ACT == ACT_SOFTPLUS) v = softplus_f(v);
            float* cp = C + (size_t)(mr_base + r) * ldc + n;
            if (ACC) *cp += v; else *cp = v;
        }
    }
}

// ---------------------------------------------------------------------------
// LayerNorm over DMODEL=512. One wave per row; wave32 shuffle reduction.
// ---------------------------------------------------------------------------
__global__ __launch_bounds__(256) void layernorm_k(
    const float* __restrict__ x, const float* __restrict__ w,
    const float* __restrict__ b, float* __restrict__ out)
{
    const int row  = blockIdx.x * 8 + (threadIdx.x >> 5);
    const int lane = threadIdx.x & 31;
    const float* xr = x + (size_t)row * DMODEL;

    float vals[16];
    float s = 0.f;
#pragma unroll
    for (int i = 0; i < 16; ++i) { vals[i] = xr[lane + 32 * i]; s += vals[i]; }
#pragma unroll
    for (int off = 16; off; off >>= 1) s += __shfl_xor(s, off, 32);
    const float mu = s * (1.f / DMODEL);

    float vs = 0.f;
#pragma unroll
    for (int i = 0; i < 16; ++i) { const float d = vals[i] - mu; vs += d * d; }
#pragma unroll
    for (int off = 16; off; off >>= 1) vs += __shfl_xor(vs, off, 32);
    const float inv = rsqrtf(vs * (1.f / DMODEL) + 1e-5f);

    float* orow = out + (size_t)row * DMODEL;
#pragma unroll
    for (int i = 0; i < 16; ++i) {
        const int c = lane + 32 * i;
        orow[c] = (vals[i] - mu) * inv * w[c] + b[c];
    }
}

// ---------------------------------------------------------------------------
// Causal depthwise conv1d (width 4) over time + SiLU. Coalesced over d.
// ---------------------------------------------------------------------------
__global__ __launch_bounds__(256) void conv_silu_k(
    const float* __restrict__ xz, const float* __restrict__ cw,
    const float* __restrict__ cb, float* __restrict__ u)
{
    const int d   = blockIdx.x * 256 + threadIdx.x;   // 0..1023
    const int row = blockIdx.y;                       // 0..4095 = b*T + t
    const int t   = row & (TLEN - 1);

    float acc = cb[d];
#pragma unroll
    for (int j = 0; j < DCONV; ++j) {
        const int tt = t + j - (DCONV - 1);
        if (tt >= 0)
            acc += cw[d * DCONV + j] *
                   xz[(size_t)(row + j - (DCONV - 1)) * (2 * DINNER) + d];
    }
    u[(size_t)row * DINNER + d] = silu_f(acc);
}

// ---------------------------------------------------------------------------
// Selective scan. One thread per (b,d) channel: 32 states + A in registers,
// B_t/C_t staged in LDS per 32-step chunk. exp via exp2 (log2e folded into A).
// Fused D-skip + silu(z) gating.
// ---------------------------------------------------------------------------
#define SCAN_BLK 128
#define SCAN_TCH 32

__global__ __launch_bounds__(SCAN_BLK) void scan_k(
    const float* __restrict__ U,      // (4096, DINNER)
    const float* __restrict__ DEL,    // (4096, DINNER)
    const float* __restrict__ XD,     // (4096, 96): dt|B|C
    const float* __restrict__ XZ,     // (4096, 2*DINNER): z at col DINNER+d
    const float* __restrict__ A_log,  // (DINNER, DSTATE)
    const float* __restrict__ Dp,     // (DINNER)
    float* __restrict__ Y)            // (4096, DINNER)
{
    __shared__ float sB[SCAN_TCH][DSTATE];
    __shared__ float sC[SCAN_TCH][DSTATE];

    const int groups = DINNER / SCAN_BLK;            // 8
    const int b  = blockIdx.x / groups;
    const int dg = blockIdx.x % groups;
    const int d  = dg * SCAN_BLK + threadIdx.x;

    const float LOG2E = 1.4426950408889634f;
    float Av[DSTATE], h[DSTATE];
#pragma unroll
    for (int n = 0; n < DSTATE; ++n) {
        Av[n] = -__expf(A_log[(size_t)d * DSTATE + n]) * LOG2E;
        h[n] = 0.f;
    }
    const float Dd = Dp[d];
    const size_t rowbase = (size_t)b * TLEN;

    for (int t0 = 0; t0 < TLEN; t0 += SCAN_TCH) {
        __syncthreads();
        for (int i = threadIdx.x; i < SCAN_TCH * DSTATE; i += SCAN_BLK) {
            const int tt = i >> 5, n = i & 31;
            const size_t r = (rowbase + t0 + tt) * 96;
            sB[tt][n] = XD[r + DTRANK + n];
            sC[tt][n] = XD[r + DTRANK + DSTATE + n];
        }
        __syncthreads();

        for (int tt = 0; tt < SCAN_TCH; ++tt) {
            const size_t r = rowbase + t0 + tt;
            const float dt = DEL[r * DINNER + d];
            const float ut = U[r * DINNER + d];
            const float du = dt * ut;
            float y = 0.f;
#pragma unroll
            for (int n = 0; n < DSTATE; ++n) {
                const float dA = exp2f(dt * Av[n]);   // v_exp_f32
                h[n] = dA * h[n] + du * sB[tt][n];
                y += h[n] * sC[tt][n];
            }
            const float z = XZ[r * (2 * DINNER) + DINNER + d];
            Y[r * DINNER + d] = (y + ut * Dd) * silu_f(z);
        }
    }
}

// ---------------------------------------------------------------------------
// Launch
// ---------------------------------------------------------------------------
extern "C" void kernel_launch(void* const* d_in, const int* in_sizes, int n_in,
                              void* d_out, int out_size, void* d_ws, size_t ws_size,
                              hipStream_t stream)
{
    const float* x_in    = (const float*)d_in[0];
    const float* in_w    = (const float*)d_in[1];
    const float* conv_w  = (const float*)d_in[2];
    const float* conv_b  = (const float*)d_in[3];
    const float* xproj_w = (const float*)d_in[4];
    const float* dt_w    = (const float*)d_in[5];
    const float* dt_b    = (const float*)d_in[6];
    const float* A_log   = (const float*)d_in[7];
    const float* Dp      = (const float*)d_in[8];
    const float* out_w   = (const float*)d_in[9];
    const float* ln_w    = (const float*)d_in[10];
    const float* ln_b    = (const float*)d_in[11];
    const float* proj_w  = (const float*)d_in[12];
    const float* proj_b  = (const float*)d_in[13];

    float* X   = (float*)d_ws;                       // (4096, 512) residual
    float* H   = X   + (size_t)NROWS * DMODEL;       // (4096, 512)
    float* XZ  = H   + (size_t)NROWS * DMODEL;       // (4096, 2048)
    float* U   = XZ  + (size_t)NROWS * 2 * DINNER;   // (4096, 1024)
    float* XD  = U   + (size_t)NROWS * DINNER;       // (4096, 96)
    float* DEL = XD  + (size_t)NROWS * 96;           // (4096, 1024)
    float* Y   = DEL + (size_t)NROWS * DINNER;       // (4096, 1024)

    hipMemcpyAsync(X, x_in, (size_t)NROWS * DMODEL * sizeof(float),
                   hipMemcpyDeviceToDevice, stream);

    for (int l = 0; l < NLAYERS; ++l) {
        layernorm_k<<<dim3(NROWS / 8), 256, 0, stream>>>(
            X, ln_w + l * DMODEL, ln_b + l * DMODEL, H);

        // xz = h @ in_w          (4096 x 2048, K=512)
        gemm_wmma_f32<4, ACT_NONE, false><<<dim3(2 * DINNER / 64, NROWS / 64), 128, 0, stream>>>(
            H, DMODEL, in_w + (size_t)l * DMODEL * 2 * DINNER, 2 * DINNER,
            nullptr, XZ, 2 * DINNER, DMODEL);

        conv_silu_k<<<dim3(DINNER / 256, NROWS), 256, 0, stream>>>(
            XZ, conv_w + (size_t)l * DINNER * DCONV, conv_b + (size_t)l * DINNER, U);

        // x_dbl = u @ xproj_w    (4096 x 96, K=1024)
        gemm_wmma_f32<2, ACT_NONE, false><<<dim3(96 / 32, NROWS / 64), 128, 0, stream>>>(
            U, DINNER, xproj_w + (size_t)l * DINNER * 96, 96,
            nullptr, XD, 96, DINNER);

        // delta = softplus(x_dbl[:, :32] @ dt_w + dt_b)   (4096 x 1024, K=32)
        gemm_wmma_f32<4, ACT_SOFTPLUS, false><<<dim3(DINNER / 64, NROWS / 64), 128, 0, stream>>>(
            XD, 96, dt_w + (size_t)l * DTRANK * DINNER, DINNER,
            dt_b + (size_t)l * DINNER, DEL, DINNER, DTRANK);

        scan_k<<<dim3(BATCH * (DINNER / SCAN_BLK)), SCAN_BLK, 0, stream>>>(
            U, DEL, XD, XZ, A_log + (size_t)l * DINNER * DSTATE,
            Dp + (size_t)l * DINNER, Y);

        // x += y @ out_w         (4096 x 512, K=1024, residual accumulate)
        gemm_wmma_f32<4, ACT_NONE, true><<<dim3(DMODEL / 64, NROWS / 64), 128, 0, stream>>>(
            Y, DINNER, out_w + (size_t)l * DINNER * DMODEL, DMODEL,
            nullptr, X, DMODEL, DINNER);
    }

    // out = x @ proj_w + proj_b  (4096 x 512, K=512)
    gemm_wmma_f32<4, ACT_NONE, false><<<dim3(DMODEL / 64, NROWS / 64), 128, 0, stream>>>(
        X, DMODEL, proj_w, DMODEL, proj_b, (float*)d_out, DMODEL, DMODEL);
}